// GraphFunc_50543175139368
// MI455X (gfx1250) — compile-verified
//
#include <hip/hip_runtime.h>

#define D_Z  64
#define HID  256
#define NCLS 5

typedef float v2f __attribute__((ext_vector_type(2)));
typedef float v4f __attribute__((ext_vector_type(4)));
typedef float v8f __attribute__((ext_vector_type(8)));

// ---------------------------------------------------------------------------
// Pass 0: per-class counts
// ---------------------------------------------------------------------------
__global__ void k_count(const int* __restrict__ label, int* __restrict__ counts, int N) {
  __shared__ int sc[NCLS];
  if (threadIdx.x < NCLS) sc[threadIdx.x] = 0;
  __syncthreads();
  for (int i = blockIdx.x * blockDim.x + threadIdx.x; i < N; i += gridDim.x * blockDim.x) {
    int c = label[i];
    if (c >= 0 && c < NCLS) atomicAdd(&sc[c], 1);
  }
  __syncthreads();
  if (threadIdx.x < NCLS) atomicAdd(&counts[threadIdx.x], sc[threadIdx.x]);
}

// ---------------------------------------------------------------------------
// Pass 1: class sums xc[g][c][d] = sum over nodes of class c of x[g][n][d]
// 256 threads = 16 nodes x 16 float4-lanes per iteration; register accumulate
// per class, LDS reduce, one float atomic per (c,d) per block.
// ---------------------------------------------------------------------------
__global__ void k_class_sum(const float* __restrict__ x, const int* __restrict__ label,
                            float* __restrict__ xc, int N) {
  const int g   = blockIdx.y;
  const int tid = threadIdx.x;     // 256
  const int d4  = tid & 15;        // float4 slot within a node (64 floats = 16 x float4)
  const int sub = tid >> 4;        // 0..15: node sub-index
  const int stride = gridDim.x * 16;
  const v4f* x4 = (const v4f*)(x + (size_t)g * N * D_Z);

  v4f acc[NCLS];
#pragma unroll
  for (int c = 0; c < NCLS; ++c) acc[c] = (v4f){0.f, 0.f, 0.f, 0.f};

  for (int n = blockIdx.x * 16 + sub; n < N; n += stride) {
    v4f v = x4[(size_t)n * 16 + d4];   // RT load: keep x resident in L2 (192MB > 164MB)
    int lab = label[n];
#pragma unroll
    for (int c = 0; c < NCLS; ++c) {
      if (lab == c) acc[c] += v;
    }
  }

  __shared__ float xcs[NCLS * D_Z];
  for (int i = tid; i < NCLS * D_Z; i += blockDim.x) xcs[i] = 0.f;
  __syncthreads();
#pragma unroll
  for (int c = 0; c < NCLS; ++c) {
    atomicAdd(&xcs[c * D_Z + d4 * 4 + 0], acc[c].x);
    atomicAdd(&xcs[c * D_Z + d4 * 4 + 1], acc[c].y);
    atomicAdd(&xcs[c * D_Z + d4 * 4 + 2], acc[c].z);
    atomicAdd(&xcs[c * D_Z + d4 * 4 + 3], acc[c].w);
  }
  __syncthreads();
  for (int i = tid; i < NCLS * D_Z; i += blockDim.x)
    atomicAdd(&xc[g * (NCLS * D_Z) + i], xcs[i]);
}

// ---------------------------------------------------------------------------
// Pass 2: per-g tiny GEMM chain with V_WMMA_F32_16X16X4_F32.
//   A1 = xc/counts, padded to 16 rows  ([16 x 64])
//   HC = relu(A1 @ W1 + b1)            ([16 x 256], rows >= 5 unused)
//   oc = HC @ W2 + b2                  (rows < 5 stored)
// One workgroup per g, 16 waves; each wave owns one 16-wide N tile.
// Fragment layouts per ISA 7.12.2: A lane->M, K = vgpr + 2*(lane>=16);
// B lane->N, same K rule; C/D vgpr r -> rows r and r+8.
// ---------------------------------------------------------------------------
__global__ void __launch_bounds__(512) k_gcn(
    const float* __restrict__ xc, const int* __restrict__ counts,
    const float* __restrict__ W1, const float* __restrict__ b1,
    const float* __restrict__ W2, const float* __restrict__ b2,
    float* __restrict__ oc) {
  __shared__ float A1[16 * D_Z];   // 4 KB
  __shared__ float HC[16 * HID];   // 16 KB
  const int g   = blockIdx.x;
  const int tid = threadIdx.x;

  for (int i = tid; i < 16 * D_Z; i += blockDim.x) {
    int c = i >> 6, d = i & 63;
    float v = 0.f;
    if (c < NCLS) {
      int cnt = counts[c];
      if (cnt < 1) cnt = 1;
      v = xc[(g * NCLS + c) * D_Z + d] / (float)cnt;
    }
    A1[i] = v;
  }
  __syncthreads();

  const int wave = tid >> 5;   // 0..15
  const int lane = tid & 31;
  const int m    = lane & 15;  // A-row / B-col within tile
  const int h    = lane >> 4;  // lane half: adds 2 to K, 8 to output row

  // ---- GEMM1: [16 x 64] @ [64 x 256] -> HC ----
  {
    const int n = wave * 16 + m;
    v8f acc = {0.f, 0.f, 0.f, 0.f, 0.f, 0.f, 0.f, 0.f};
#pragma unroll
    for (int k0 = 0; k0 < D_Z; k0 += 4) {
      v2f a, b;
      a[0] = A1[m * D_Z + k0 + 2 * h + 0];
      a[1] = A1[m * D_Z + k0 + 2 * h + 1];
      b[0] = W1[(k0 + 2 * h + 0) * HID + n];
      b[1] = W1[(k0 + 2 * h + 1) * HID + n];
      acc = __builtin_amdgcn_wmma_f32_16x16x4_f32(false, a, false, b,
                                                  (short)0, acc, false, false);
    }
    float bias = b1[n];
#pragma unroll
    for (int r = 0; r < 8; ++r) {
      int row = r + 8 * h;
      float v = acc[r] + bias;
      HC[row * HID + n] = v > 0.f ? v : 0.f;
    }
  }
  __syncthreads();

  // ---- GEMM2: [16 x 256] @ [256 x 64] -> oc ---- (wave-granular branch: EXEC all-1s)
  if (wave < 4) {
    const int n = wave * 16 + m;
    v8f acc = {0.f, 0.f, 0.f, 0.f, 0.f, 0.f, 0.f, 0.f};
#pragma unroll 8
    for (int k0 = 0; k0 < HID; k0 += 4) {
      v2f a, b;
      a[0] = HC[m * HID + k0 + 2 * h + 0];
      a[1] = HC[m * HID + k0 + 2 * h + 1];
      b[0] = W2[(k0 + 2 * h + 0) * D_Z + n];
      b[1] = W2[(k0 + 2 * h + 1) * D_Z + n];
      acc = __builtin_amdgcn_wmma_f32_16x16x4_f32(false, a, false, b,
                                                  (short)0, acc, false, false);
    }
    float bias = b2[n];
#pragma unroll
    for (int r = 0; r < 8; ++r) {
      int row = r + 8 * h;
      if (row < NCLS) oc[(g * NCLS + row) * D_Z + n] = acc[r] + bias;
    }
  }
}

// ---------------------------------------------------------------------------
// Pass 3: out[g][n][:] = x[g][n][:] + oc[g][label[n]][:]
// x re-read hits L2 (RT); output streamed with non-temporal stores so it
// doesn't evict x's lines.
// ---------------------------------------------------------------------------
__global__ void k_scatter(const float* __restrict__ x, const int* __restrict__ label,
                          const float* __restrict__ oc, float* __restrict__ out, int N) {
  const int g = blockIdx.y;
  const int i = blockIdx.x * blockDim.x + threadIdx.x;   // over N*16 float4s
  if (i >= N * 16) return;
  const int n   = i >> 4;
  const int d4  = i & 15;
  const int lab = label[n];

  const v4f* x4  = (const v4f*)x;
  const v4f* oc4 = (const v4f*)oc;
  v4f*       o4  = (v4f*)out;

  size_t idx = (size_t)g * N * 16 + i;
  v4f xv = x4[idx];
  v4f av = oc4[(g * NCLS + lab) * 16 + d4];
  v4f ov = xv + av;
  __builtin_nontemporal_store(ov, o4 + idx);
}

// ---------------------------------------------------------------------------
extern "C" void kernel_launch(void* const* d_in, const int* in_sizes, int n_in,
                              void* d_out, int out_size, void* d_ws, size_t ws_size,
                              hipStream_t stream) {
  const float* x   = (const float*)d_in[0];
  const int*   lab = (const int*)d_in[1];
  const float* W1  = (const float*)d_in[2];
  const float* b1  = (const float*)d_in[3];
  const float* W2  = (const float*)d_in[4];
  const float* b2  = (const float*)d_in[5];
  float* out = (float*)d_out;

  const int N = in_sizes[1];
  const int G = in_sizes[0] / (N * D_Z);

  char*  ws     = (char*)d_ws;
  int*   counts = (int*)ws;                                              // NCLS ints (in 256B slot)
  float* xc     = (float*)(ws + 256);                                    // G*NCLS*64 floats
  float* oc     = (float*)(ws + 256 + (size_t)G * NCLS * D_Z * sizeof(float)); // 16B-aligned

  // zero counts + xc accumulators (workspace is not re-zeroed between replays)
  hipMemsetAsync(ws, 0, 256 + (size_t)G * NCLS * D_Z * sizeof(float), stream);

  k_count<<<40, 256, 0, stream>>>(lab, counts, N);

  dim3 gs(64, G);
  k_class_sum<<<gs, 256, 0, stream>>>(x, lab, xc, N);

  k_gcn<<<G, 512, 0, stream>>>(xc, counts, W1, b1, W2, b2, oc);

  dim3 sg((N * 16 + 255) / 256, G);
  k_scatter<<<sg, 256, 0, stream>>>(x, lab, oc, out, N);
}